// SubArrayGSMBeamformingNet_87385404605127
// MI455X (gfx1250) — compile-verified
//
#include <hip/hip_runtime.h>
#include <math.h>

// ---------------------------------------------------------------------------
// SubArrayGSMBeamformingNet for MI455X (gfx1250, wave32, WMMA).
//
// Algebra: V_real^2 + V_imag^2 = Dr^2 + Di^2 (cos^2+sin^2 = 1), and each
// selected set maps Nk=8 antennas onto exactly one RF chain, so
//   V_F[b,c] = sqrt(8 * (||Dr[b,c]||_F^2 + ||Di[b,c]||_F^2)).
// => no (B,CS,Nt,Nrf) tensors materialized. Output (42 MB) dominates: HBM
// floor ~1.8 us @ 23.3 TB/s; GEMMs (~2.2 GFLOP, exact-f32 WMMA 16x16x4) are
// free by comparison. Dr-GEMM + Di-GEMM + normalization are fused into one
// kernel (shared A fragments, in-register 16-lane norm reduction), which
// eliminates a read-modify-write pass over 8 MB of Dr/Di.
// ---------------------------------------------------------------------------

typedef __attribute__((ext_vector_type(2))) float v2f;
typedef __attribute__((ext_vector_type(4))) float v4f;
typedef __attribute__((ext_vector_type(8))) float v8f;

constexpr int BATCH    = 1024;
constexpr int FEAT     = 512;
constexpr int NT       = 64;
constexpr int NRF      = 4;
constexpr int CSETSIZE = 64;
constexpr int DDIM     = CSETSIZE * NRF * NRF;   // 1024

constexpr size_t SZ_A    = (size_t)BATCH * NT * NT;      // 4,194,304
constexpr size_t SZ_D    = (size_t)BATCH * DDIM;         // 1,048,576
constexpr size_t SZ_CSET = (size_t)CSETSIZE * NT * NRF;  // 16,384
constexpr size_t OFF_AIMAG = SZ_A;
constexpr size_t OFF_DR    = 2 * SZ_A;
constexpr size_t OFF_DI    = OFF_DR + SZ_D;
constexpr size_t OFF_CSET  = OFF_DI + SZ_D;

// ---- 1) zero-fill A_real/A_imag (streaming, non-temporal b128 stores) -----
__global__ __launch_bounds__(256) void zero_fill4(v4f* __restrict__ p, size_t n4) {
    size_t i = (size_t)blockIdx.x * blockDim.x + threadIdx.x;
    size_t stride = (size_t)gridDim.x * blockDim.x;
    const v4f z = {0.f, 0.f, 0.f, 0.f};
    for (; i < n4; i += stride)
        __builtin_nontemporal_store(z, p + i);   // TH=NT: never re-read here
}

// ---- 2) theta GEMM: theta = x @ W_A^T + b_A, cos/sin onto A diagonals -----
// One wave -> one 16x16 theta tile via V_WMMA_F32_16X16X4_F32 (exact f32).
// A frag (ISA layout): lane<16 holds K=0,1; lane>=16 holds K=2,3.
__global__ __launch_bounds__(128) void theta_gemm_diag(
    const float* __restrict__ X,   // (BATCH, FEAT)
    const float* __restrict__ WA,  // (NT, FEAT)
    const float* __restrict__ bA,  // (NT)
    float* __restrict__ Areal,     // (BATCH, NT, NT) pre-zeroed
    float* __restrict__ Aimag)
{
    const int lane  = threadIdx.x;
    const int half  = lane >> 4;
    const int l16   = lane & 15;
    const int mTile = blockIdx.y * 64 + threadIdx.y * 16;
    const int nTile = blockIdx.x * 16;

    const float* xrow = X  + (size_t)(mTile + l16) * FEAT;
    const float* wrow = WA + (size_t)(nTile + l16) * FEAT;

    v8f acc = {0.f, 0.f, 0.f, 0.f, 0.f, 0.f, 0.f, 0.f};
#pragma unroll 8
    for (int k = 0; k < FEAT; k += 4) {
        v2f a = *(const v2f*)(xrow + k + half * 2);
        v2f b = *(const v2f*)(wrow + k + half * 2);
        acc = __builtin_amdgcn_wmma_f32_16x16x4_f32(
            false, a, false, b, (short)0, acc, false, false);
    }

    const int n = nTile + l16;               // antenna index (C/D: N = lane%16)
    const float bv = bA[n];
#pragma unroll
    for (int r = 0; r < 8; ++r) {            // C/D: VGPR r, lane>=16 -> M+8
        const int m = mTile + r + half * 8;  // batch index
        const float th = acc[r] + bv;
        float s, c;
        sincosf(th, &s, &c);
        const size_t diag = (size_t)m * (NT * NT) + (size_t)n * NT + n;
        Areal[diag] = c;
        Aimag[diag] = s;
    }
}

// ---- 3) fused D GEMMs + normalization -------------------------------------
// One wave computes matching 16x16 tiles of Dr AND Di (shared A fragments,
// 2 WMMAs per k-step). Tile columns n = nTile..nTile+15 are exactly one
// codebook entry c = nTile/16, so the (b,c) Frobenius norm is a 16-lane
// reduction over one accumulator VGPR: V_F = sqrt(8*sum(dr^2+di^2)).
// Normalized values stored directly -> no separate RMW pass.
__global__ __launch_bounds__(128) void d_gemm_norm(
    const float* __restrict__ X,    // (BATCH, FEAT)
    const float* __restrict__ Wr,   // (DDIM, FEAT)
    const float* __restrict__ br,   // (DDIM)
    const float* __restrict__ Wi,   // (DDIM, FEAT)
    const float* __restrict__ bi,   // (DDIM)
    float* __restrict__ Dr,         // (BATCH, DDIM)
    float* __restrict__ Di)
{
    const int lane  = threadIdx.x;
    const int half  = lane >> 4;
    const int l16   = lane & 15;
    const int mTile = blockIdx.y * 64 + threadIdx.y * 16;
    const int nTile = blockIdx.x * 16;

    const float* xrow  = X  + (size_t)(mTile + l16) * FEAT;
    const float* wrrow = Wr + (size_t)(nTile + l16) * FEAT;
    const float* wirow = Wi + (size_t)(nTile + l16) * FEAT;

    v8f accR = {0.f, 0.f, 0.f, 0.f, 0.f, 0.f, 0.f, 0.f};
    v8f accI = {0.f, 0.f, 0.f, 0.f, 0.f, 0.f, 0.f, 0.f};
#pragma unroll 4
    for (int k = 0; k < FEAT; k += 4) {
        v2f a  = *(const v2f*)(xrow  + k + half * 2);  // shared by both GEMMs
        v2f bR = *(const v2f*)(wrrow + k + half * 2);
        v2f bI = *(const v2f*)(wirow + k + half * 2);
        accR = __builtin_amdgcn_wmma_f32_16x16x4_f32(
            false, a, false, bR, (short)0, accR, false, false);
        accI = __builtin_amdgcn_wmma_f32_16x16x4_f32(
            false, a, false, bI, (short)0, accI, false, false);
    }

    const float bvr = br[nTile + l16];
    const float bvi = bi[nTile + l16];
#pragma unroll
    for (int r = 0; r < 8; ++r) {
        const float dr = accR[r] + bvr;
        const float di = accI[r] + bvi;
        float s = dr * dr + di * di;
#pragma unroll
        for (int m = 8; m >= 1; m >>= 1)
            s += __shfl_xor(s, m, 16);       // per-(b,c) 16-lane reduction
        const float sc = 2.0f / sqrtf(8.0f * s);  // SQRT_NRF / V_F, Nk = 8
        const int m_ = mTile + r + half * 8;
        const size_t o = (size_t)m_ * DDIM + nTile + l16;
        Dr[o] = dr * sc;
        Di[o] = di * sc;
    }
}

// ---- 4) CSet passthrough copy ---------------------------------------------
__global__ __launch_bounds__(256) void copy4(
    const v4f* __restrict__ src, v4f* __restrict__ dst, int n4)
{
    int i = blockIdx.x * blockDim.x + threadIdx.x;
    if (i < n4) dst[i] = src[i];
}

extern "C" void kernel_launch(void* const* d_in, const int* in_sizes, int n_in,
                              void* d_out, int out_size, void* d_ws, size_t ws_size,
                              hipStream_t stream) {
    const float* x    = (const float*)d_in[0];
    const float* W_A  = (const float*)d_in[1];
    const float* b_A  = (const float*)d_in[2];
    const float* W_Dr = (const float*)d_in[3];
    const float* b_Dr = (const float*)d_in[4];
    const float* W_Di = (const float*)d_in[5];
    const float* b_Di = (const float*)d_in[6];
    const float* CSet = (const float*)d_in[7];

    float* out   = (float*)d_out;
    float* Areal = out;
    float* Aimag = out + OFF_AIMAG;
    float* Dr    = out + OFF_DR;
    float* Di    = out + OFF_DI;
    float* Cout  = out + OFF_CSET;

    // 1) zero A_real + A_imag (contiguous 2*SZ_A floats, streaming NT stores)
    const size_t n4 = (2 * SZ_A) / 4;       // 2,097,152 float4s
    zero_fill4<<<dim3((unsigned)((n4 + 255) / 256)), dim3(256), 0, stream>>>(
        (v4f*)out, n4);

    // 2) theta GEMM -> cos/sin on diagonals (stream-ordered after zero-fill)
    dim3 gblk(32, 4);                       // 4 wave32s, one 16x16 tile each
    theta_gemm_diag<<<dim3(NT / 16, BATCH / 64), gblk, 0, stream>>>(
        x, W_A, b_A, Areal, Aimag);

    // 3) fused Dr+Di GEMMs with in-register normalization
    d_gemm_norm<<<dim3(DDIM / 16, BATCH / 64), gblk, 0, stream>>>(
        x, W_Dr, b_Dr, W_Di, b_Di, Dr, Di);

    // 4) CSet copy
    copy4<<<dim3((unsigned)(SZ_CSET / 4 / 256)), dim3(256), 0, stream>>>(
        (const v4f*)CSet, (v4f*)Cout, (int)(SZ_CSET / 4));
}